// InvariantPointAttention_35716948033668
// MI455X (gfx1250) — compile-verified
//
#include <hip/hip_runtime.h>
#include <hip/hip_bf16.h>
#include <math.h>

#define N   512
#define DIM 384
#define H   8
#define SK  16
#define SV  16
#define PK  4
#define PV  4
#define PAIR 128
#define FEAT 1280   // H*(SV + PV*3 + PV + PAIR) = 8*160

typedef float v2f __attribute__((ext_vector_type(2)));
typedef float v8f __attribute__((ext_vector_type(8)));

__device__ __forceinline__ v8f wmma4(v2f a, v2f b, v8f c) {
    // D(16x16,f32) = A(16x4,f32) * B(4x16,f32) + C
    return __builtin_amdgcn_wmma_f32_16x16x4_f32(false, a, false, b, (short)0, c,
                                                 false, false);
}

// ---------------------------------------------------------------------------
// Generic WMMA f32 GEMM: C[M,Ncols] = A[M,K] * B[K,Ncols] (+ bias[Ncols])
// One wave (32 threads) per 16x16 output tile. K must be a multiple of 4.
// ---------------------------------------------------------------------------
__global__ __launch_bounds__(32)
void gemm_wmma_f32(const float* __restrict__ A, int lda,
                   const float* __restrict__ B, int ldb,
                   float* __restrict__ C, int ldc,
                   int K, const float* __restrict__ bias) {
    const int lane = threadIdx.x;
    const int half = lane >> 4;      // 0: K+0/1 ; 1: K+2/3
    const int l15  = lane & 15;
    const int colBase = blockIdx.x * 16;
    const int rowBase = blockIdx.y * 16;

    v8f acc = {};
    #pragma unroll 4
    for (int k = 0; k < K; k += 4) {
        v2f a, b;
        const float* ap = A + (rowBase + l15) * lda + k + half * 2;
        a.x = ap[0]; a.y = ap[1];
        const float* bp = B + (k + half * 2) * ldb + colBase + l15;
        b.x = bp[0]; b.y = bp[ldb];
        acc = wmma4(a, b, acc);
    }
    #pragma unroll
    for (int r = 0; r < 8; ++r) {
        int row = rowBase + r + half * 8;
        int col = colBase + l15;
        float v = acc[r];
        if (bias) v += bias[col];
        C[row * ldc + col] = v;
    }
}

// ---------------------------------------------------------------------------
// Rotate local-frame points to global frame; emit per-head (H,N,12) layouts
// and squared norms for q/k.  p_g[r] = sum_c p_l[c]*R[c][r] + t[r]
// ---------------------------------------------------------------------------
__global__ void points_to_global(const float* __restrict__ qpl,
                                 const float* __restrict__ kpl,
                                 const float* __restrict__ vpl,
                                 const float* __restrict__ rot,
                                 const float* __restrict__ trans,
                                 float* __restrict__ qpg,
                                 float* __restrict__ kpg,
                                 float* __restrict__ vpg,
                                 float* __restrict__ qn,
                                 float* __restrict__ kn) {
    int idx = blockIdx.x * blockDim.x + threadIdx.x;  // n*H + h
    if (idx >= N * H) return;
    int n = idx / H, h = idx % H;
    const float* R = rot + n * 9;  // R[c][r] = R[c*3+r]
    float t0 = trans[n * 3 + 0], t1 = trans[n * 3 + 1], t2 = trans[n * 3 + 2];
    float sq = 0.f, sk = 0.f;
    #pragma unroll
    for (int d = 0; d < PK; ++d) {
        {   const float* p = qpl + n * (H * PK * 3) + h * (PK * 3) + d * 3;
            float gx = p[0]*R[0] + p[1]*R[3] + p[2]*R[6] + t0;
            float gy = p[0]*R[1] + p[1]*R[4] + p[2]*R[7] + t1;
            float gz = p[0]*R[2] + p[1]*R[5] + p[2]*R[8] + t2;
            float* o = qpg + h * (N * 12) + n * 12 + d * 3;
            o[0] = gx; o[1] = gy; o[2] = gz;
            sq += gx*gx + gy*gy + gz*gz; }
        {   const float* p = kpl + n * (H * PK * 3) + h * (PK * 3) + d * 3;
            float gx = p[0]*R[0] + p[1]*R[3] + p[2]*R[6] + t0;
            float gy = p[0]*R[1] + p[1]*R[4] + p[2]*R[7] + t1;
            float gz = p[0]*R[2] + p[1]*R[5] + p[2]*R[8] + t2;
            float* o = kpg + h * (N * 12) + n * 12 + d * 3;
            o[0] = gx; o[1] = gy; o[2] = gz;
            sk += gx*gx + gy*gy + gz*gz; }
        {   const float* p = vpl + n * (H * PV * 3) + h * (PV * 3) + d * 3;
            float gx = p[0]*R[0] + p[1]*R[3] + p[2]*R[6] + t0;
            float gy = p[0]*R[1] + p[1]*R[4] + p[2]*R[7] + t1;
            float gz = p[0]*R[2] + p[1]*R[5] + p[2]*R[8] + t2;
            float* o = vpg + h * (N * 12) + n * 12 + d * 3;
            o[0] = gx; o[1] = gy; o[2] = gz; }
    }
    qn[h * N + n] = sq;
    kn[h * N + n] = sk;
}

// ---------------------------------------------------------------------------
// Pair bias: bias[h][i][j] = pair_scale * (pairwise[i,j,:]@W_pair[:,h] + b_pair[h])
// GEMM over flattened (i,j) rows: (N*N,128)@(128,8 pad 16). One wave / 16 rows.
// ---------------------------------------------------------------------------
__global__ __launch_bounds__(32)
void pair_bias_wmma(const float* __restrict__ pair,
                    const float* __restrict__ Wp,   // (128,8)
                    const float* __restrict__ bp,   // (8)
                    float* __restrict__ bias) {     // (H,N,N)
    const int lane = threadIdx.x;
    const int half = lane >> 4, l15 = lane & 15;
    const int rowBase = blockIdx.x * 16;  // flattened ij
    v8f acc = {};
    #pragma unroll 4
    for (int k = 0; k < PAIR; k += 4) {
        v2f a, b;
        const float* ap = pair + (long)(rowBase + l15) * PAIR + k + half * 2;
        a.x = ap[0]; a.y = ap[1];
        int kk = k + half * 2;
        b.x = (l15 < H) ? Wp[kk * H + l15]       : 0.f;
        b.y = (l15 < H) ? Wp[(kk + 1) * H + l15] : 0.f;
        acc = wmma4(a, b, acc);
    }
    const float pair_scale = 0.57735026918962576f;  // 3^-0.5
    if (l15 < H) {
        float add = bp[l15];
        #pragma unroll
        for (int r = 0; r < 8; ++r) {
            int flat = rowBase + r + half * 8;
            bias[l15 * (N * N) + flat] = pair_scale * (acc[r] + add);
        }
    }
}

// ---------------------------------------------------------------------------
// Logits + softmax, one wave per (head, 16-row i tile).
// scalar qk via WMMA K=16, point dot via WMMA K=12; softmax over j in LDS.
// ---------------------------------------------------------------------------
__global__ __launch_bounds__(32)
void logits_softmax(const float* __restrict__ qs, const float* __restrict__ ks,
                    const float* __restrict__ qpg, const float* __restrict__ kpg,
                    const float* __restrict__ qn, const float* __restrict__ kn,
                    const float* __restrict__ bias,   // (H,N,N)
                    const float* __restrict__ pwv,    // point_weights (H)
                    float* __restrict__ attn) {       // (H,N,N)
    __shared__ float sm[16][N];
    const int lane = threadIdx.x, half = lane >> 4, l15 = lane & 15;
    const int iBase = blockIdx.x * 16;
    const int h = blockIdx.y;
    const float scalar_scale = 0.144337567297406f;  // (3*16)^-0.5
    const float point_scale  = 0.136082763487954f;  // (12*4.5)^-0.5
    const float pw = log1pf(expf(pwv[h]));

    // A fragments (held across the whole j loop)
    v2f aq[4], ap[3];
    #pragma unroll
    for (int t = 0; t < 4; ++t) {
        const float* p = qs + (iBase + l15) * (H * SK) + h * SK + t * 4 + half * 2;
        aq[t].x = p[0]; aq[t].y = p[1];
    }
    #pragma unroll
    for (int t = 0; t < 3; ++t) {
        const float* p = qpg + h * (N * 12) + (iBase + l15) * 12 + t * 4 + half * 2;
        ap[t].x = p[0]; ap[t].y = p[1];
    }
    float qnr[8];
    #pragma unroll
    for (int r = 0; r < 8; ++r) qnr[r] = qn[h * N + iBase + r + half * 8];

    for (int jt = 0; jt < N / 16; ++jt) {
        const int jBase = jt * 16;
        v8f accS = {}, accP = {};
        #pragma unroll
        for (int t = 0; t < 4; ++t) {
            v2f b;
            const float* p = ks + (jBase + l15) * (H * SK) + h * SK + t * 4 + half * 2;
            b.x = p[0]; b.y = p[1];
            accS = wmma4(aq[t], b, accS);
        }
        #pragma unroll
        for (int t = 0; t < 3; ++t) {
            v2f b;
            const float* p = kpg + h * (N * 12) + (jBase + l15) * 12 + t * 4 + half * 2;
            b.x = p[0]; b.y = p[1];
            accP = wmma4(ap[t], b, accP);
        }
        float knj = kn[h * N + jBase + l15];
        #pragma unroll
        for (int r = 0; r < 8; ++r) {
            int i = iBase + r + half * 8;
            float l = accS[r] * scalar_scale
                    + bias[h * (N * N) + i * N + jBase + l15]
                    - 0.5f * point_scale * pw * (qnr[r] + knj - 2.f * accP[r]);
            sm[r + half * 8][jBase + l15] = l;
        }
    }
    __syncthreads();

    // per-row softmax over 512, whole wave cooperates per row
    for (int r2 = 0; r2 < 16; ++r2) {
        float m = -INFINITY;
        #pragma unroll
        for (int t = 0; t < N / 32; ++t) m = fmaxf(m, sm[r2][lane + t * 32]);
        #pragma unroll
        for (int off = 16; off > 0; off >>= 1) m = fmaxf(m, __shfl_xor(m, off, 32));
        float s = 0.f;
        #pragma unroll
        for (int t = 0; t < N / 32; ++t) s += expf(sm[r2][lane + t * 32] - m);
        #pragma unroll
        for (int off = 16; off > 0; off >>= 1) s += __shfl_xor(s, off, 32);
        float inv = 1.f / s;
        int i = iBase + r2;
        #pragma unroll
        for (int t = 0; t < N / 32; ++t) {
            int j = lane + t * 32;
            attn[h * (N * N) + i * N + j] = expf(sm[r2][j] - m) * inv;
        }
    }
}

// ---------------------------------------------------------------------------
// out_s = attn @ v_s  and  pts = attn @ v_p (shared A fragments), K = 512.
// ---------------------------------------------------------------------------
__global__ __launch_bounds__(32)
void attn_sv(const float* __restrict__ attn, const float* __restrict__ vs,
             const float* __restrict__ vpg,
             float* __restrict__ feats, float* __restrict__ pts) {
    const int lane = threadIdx.x, half = lane >> 4, l15 = lane & 15;
    const int iBase = blockIdx.x * 16;
    const int h = blockIdx.y;
    v8f accS = {}, accP = {};
    #pragma unroll 4
    for (int k = 0; k < N; k += 4) {
        v2f a;
        const float* apq = attn + h * (N * N) + (iBase + l15) * N + k + half * 2;
        a.x = apq[0]; a.y = apq[1];
        int kk = k + half * 2;
        v2f bs;
        const float* bp = vs + kk * (H * SV) + h * SV + l15;
        bs.x = bp[0]; bs.y = bp[H * SV];
        accS = wmma4(a, bs, accS);
        v2f bv;
        const float* bq = vpg + h * (N * 12) + kk * 12 + l15;
        bv.x = (l15 < 12) ? bq[0]  : 0.f;
        bv.y = (l15 < 12) ? bq[12] : 0.f;
        accP = wmma4(a, bv, accP);
    }
    #pragma unroll
    for (int r = 0; r < 8; ++r) {
        int i = iBase + r + half * 8;
        feats[i * FEAT + h * SV + l15] = accS[r];
        if (l15 < 12) pts[i * (H * 12) + h * 12 + l15] = accP[r];
    }
}

// ---------------------------------------------------------------------------
// out_pair: per token i, (8heads pad16, 512) @ pairwise[i](512,128). One wave
// per (i, 16-col p tile); streams pairwise_repr a second time.
// ---------------------------------------------------------------------------
__global__ __launch_bounds__(32)
void attn_pair(const float* __restrict__ attn, const float* __restrict__ pair,
               float* __restrict__ feats) {
    const int lane = threadIdx.x, half = lane >> 4, l15 = lane & 15;
    const int i = blockIdx.y;
    const int pBase = blockIdx.x * 16;
    v8f acc = {};
    #pragma unroll 4
    for (int k = 0; k < N; k += 4) {
        v2f a;
        int ko = k + half * 2;
        a.x = (l15 < H) ? attn[l15 * (N * N) + i * N + ko]     : 0.f;
        a.y = (l15 < H) ? attn[l15 * (N * N) + i * N + ko + 1] : 0.f;
        v2f b;
        const float* bp = pair + ((long)i * N + ko) * PAIR + pBase + l15;
        b.x = bp[0]; b.y = bp[PAIR];
        acc = wmma4(a, b, acc);
    }
    if (half == 0) {   // rows 0..7 hold heads 0..7
        #pragma unroll
        for (int r = 0; r < 8; ++r)
            feats[i * FEAT + 256 + r * PAIR + pBase + l15] = acc[r];
    }
}

// ---------------------------------------------------------------------------
// pts_local = (pts - t) @ R^T ; norms; write into feats concat layout.
// ---------------------------------------------------------------------------
__global__ void pts_local_kernel(const float* __restrict__ pts,
                                 const float* __restrict__ rot,
                                 const float* __restrict__ trans,
                                 float* __restrict__ feats) {
    int idx = blockIdx.x * blockDim.x + threadIdx.x;  // n*H*PV + h*PV + d
    if (idx >= N * H * PV) return;
    int n = idx / (H * PV);
    int rem = idx % (H * PV);
    int h = rem / PV, d = rem % PV;
    const float* p = pts + n * (H * 12) + h * 12 + d * 3;
    float x = p[0] - trans[n * 3 + 0];
    float y = p[1] - trans[n * 3 + 1];
    float z = p[2] - trans[n * 3 + 2];
    const float* R = rot + n * 9;  // R[r][c] = R[r*3+c]
    float lx = x * R[0] + y * R[1] + z * R[2];
    float ly = x * R[3] + y * R[4] + z * R[5];
    float lz = x * R[6] + y * R[7] + z * R[8];
    float* o = feats + n * FEAT + 128 + h * 12 + d * 3;
    o[0] = lx; o[1] = ly; o[2] = lz;
    feats[n * FEAT + 224 + h * PV + d] = sqrtf(lx * lx + ly * ly + lz * lz + 1e-8f);
}

// ---------------------------------------------------------------------------
extern "C" void kernel_launch(void* const* d_in, const int* in_sizes, int n_in,
                              void* d_out, int out_size, void* d_ws, size_t ws_size,
                              hipStream_t stream) {
    (void)in_sizes; (void)n_in; (void)out_size; (void)ws_size;
    const float* x     = (const float*)d_in[0];
    const float* pair  = (const float*)d_in[1];
    const float* rot   = (const float*)d_in[2];
    const float* trans = (const float*)d_in[3];
    const float* Wsq   = (const float*)d_in[4];
    const float* Wsk   = (const float*)d_in[5];
    const float* Wsv   = (const float*)d_in[6];
    const float* Wpq   = (const float*)d_in[7];
    const float* Wpk   = (const float*)d_in[8];
    const float* Wpv   = (const float*)d_in[9];
    const float* Wpair = (const float*)d_in[10];
    const float* bpair = (const float*)d_in[11];
    const float* pwv   = (const float*)d_in[12];
    const float* Wout  = (const float*)d_in[13];
    const float* bout  = (const float*)d_in[14];
    float* out = (float*)d_out;

    float* ws = (float*)d_ws;
    size_t o = 0;
    float* qs   = ws + o; o += (size_t)N * H * SK;          // 65536
    float* ks   = ws + o; o += (size_t)N * H * SK;
    float* vs   = ws + o; o += (size_t)N * H * SV;
    float* qpl  = ws + o; o += (size_t)N * H * PK * 3;      // 49152
    float* kpl  = ws + o; o += (size_t)N * H * PK * 3;
    float* vpl  = ws + o; o += (size_t)N * H * PV * 3;
    float* qpg  = ws + o; o += (size_t)H * N * 12;
    float* kpg  = ws + o; o += (size_t)H * N * 12;
    float* vpg  = ws + o; o += (size_t)H * N * 12;
    float* qn   = ws + o; o += (size_t)H * N;
    float* kn   = ws + o; o += (size_t)H * N;
    float* bias = ws + o; o += (size_t)H * N * N;           // 8 MB
    float* attn = ws + o; o += (size_t)H * N * N;           // 8 MB
    float* pts  = ws + o; o += (size_t)N * H * 12;
    float* feat = ws + o; o += (size_t)N * FEAT;            // 2.6 MB

    dim3 wave(32);

    // projections: x(512,384) @ W
    gemm_wmma_f32<<<dim3((H*SK)/16,   N/16), wave, 0, stream>>>(x, DIM, Wsq, H*SK,   qs,  H*SK,   DIM, nullptr);
    gemm_wmma_f32<<<dim3((H*SK)/16,   N/16), wave, 0, stream>>>(x, DIM, Wsk, H*SK,   ks,  H*SK,   DIM, nullptr);
    gemm_wmma_f32<<<dim3((H*SV)/16,   N/16), wave, 0, stream>>>(x, DIM, Wsv, H*SV,   vs,  H*SV,   DIM, nullptr);
    gemm_wmma_f32<<<dim3((H*PK*3)/16, N/16), wave, 0, stream>>>(x, DIM, Wpq, H*PK*3, qpl, H*PK*3, DIM, nullptr);
    gemm_wmma_f32<<<dim3((H*PK*3)/16, N/16), wave, 0, stream>>>(x, DIM, Wpk, H*PK*3, kpl, H*PK*3, DIM, nullptr);
    gemm_wmma_f32<<<dim3((H*PV*3)/16, N/16), wave, 0, stream>>>(x, DIM, Wpv, H*PV*3, vpl, H*PV*3, DIM, nullptr);

    points_to_global<<<(N * H + 255) / 256, 256, 0, stream>>>(
        qpl, kpl, vpl, rot, trans, qpg, kpg, vpg, qn, kn);

    pair_bias_wmma<<<(N * N) / 16, wave, 0, stream>>>(pair, Wpair, bpair, bias);

    logits_softmax<<<dim3(N / 16, H), wave, 0, stream>>>(
        qs, ks, qpg, kpg, qn, kn, bias, pwv, attn);

    attn_sv<<<dim3(N / 16, H), wave, 0, stream>>>(attn, vs, vpg, feat, pts);

    attn_pair<<<dim3(PAIR / 16, N), wave, 0, stream>>>(attn, pair, feat);

    pts_local_kernel<<<(N * H * PV + 255) / 256, 256, 0, stream>>>(pts, rot, trans, feat);

    // final: feats(512,1280) @ W_out(1280,384) + b_out
    gemm_wmma_f32<<<dim3(DIM / 16, N / 16), wave, 0, stream>>>(
        feat, FEAT, Wout, DIM, out, DIM, FEAT, bout);
}